// AVQ_19095424598550
// MI455X (gfx1250) — compile-verified
//
#include <hip/hip_runtime.h>
#include <math.h>

// Problem constants (match reference)
#define NSTAGE 4
#define KCB    1024
#define SK     4096      // NSTAGE*KCB
#define DDIM   256
#define BEAM   4
#define NPTS   4096
#define EPSF   1e-12f
#define INFF   3.0e38f

typedef float v2f __attribute__((ext_vector_type(2)));
typedef float v8f __attribute__((ext_vector_type(8)));
typedef int   v4i __attribute__((vector_size(16)));   // matches async builtin param

// LDS strides (padded so 16 rows map to distinct bank groups: stride%64==4)
#define QSTR 260   // 256 K + 4 pad
#define BSTR 68    // 64 K + 4 pad (68*4 = 272 B, keeps float4 stores 16B aligned)
#define DSTR 68    // 64 cols + 4 pad
#define BBUF (64 * BSTR)   // one btile buffer, floats

// gfx1250 async global->LDS path (ASYNCcnt), guarded so compile never breaks.
#if __has_builtin(__builtin_amdgcn_global_load_async_to_lds_b128) && \
    __has_builtin(__builtin_amdgcn_s_wait_asynccnt)
#define HAVE_ASYNC 1
#else
#define HAVE_ASYNC 0
#endif

#define AS1 __attribute__((address_space(1)))
#define AS3 __attribute__((address_space(3)))

// ---------------------------------------------------------------------------
// ||codebook_row||^2
// ---------------------------------------------------------------------------
__global__ __launch_bounds__(256)
void avq_cb2_kernel(const float* __restrict__ cb, float* __restrict__ cb2) {
  int r = blockIdx.x * blockDim.x + threadIdx.x;
  if (r >= SK) return;
  const float4* p = (const float4*)(cb + (size_t)r * DDIM);
  float s = 0.f;
#pragma unroll 8
  for (int i = 0; i < DDIM / 4; ++i) {
    float4 v = p[i];
    s += v.x * v.x + v.y * v.y + v.z * v.z + v.w * v.w;
  }
  cb2[r] = s;
}

__global__ __launch_bounds__(256)
void avq_zero_used_kernel(float* __restrict__ used) {
  int i = blockIdx.x * blockDim.x + threadIdx.x;
  if (i < SK) used[i] = 0.f;
}

// ---------------------------------------------------------------------------
// Stage one 64-col x 64-K codebook slice into an LDS buffer.
// 128 threads x 8 issues x 16 B = 16 KB.
// Async path: global_load_async_to_lds_b128, no VGPR round-trip, ASYNCcnt.
// ---------------------------------------------------------------------------
__device__ __forceinline__
void avq_stage_slice(const float* __restrict__ cb, int colBase, int ks,
                     float* __restrict__ dstBuf, int tid)
{
#pragma unroll
  for (int it = 0; it < 8; ++it) {
    const int i  = it * 128 + tid;
    const int c  = i >> 4;
    const int kq = (i & 15) << 2;
    const float* g = cb + (size_t)(colBase + c) * DDIM + ks + kq;
    float* l = dstBuf + c * BSTR + kq;
#if HAVE_ASYNC
    __builtin_amdgcn_global_load_async_to_lds_b128(
        (AS1 v4i*)g, (AS3 v4i*)l, 0, 0);
#else
    float4 v = *(const float4*)g;
    l[0] = v.x; l[1] = v.y; l[2] = v.z; l[3] = v.w;
#endif
  }
}

// ---------------------------------------------------------------------------
// Fused distance GEMM (V_WMMA_F32_16X16X4_F32) + masked per-row top-4.
//   stage0==1: queries = x[row], no ban, output tuples[row][w][0]=idx_w, rest -1
//   stage0==0: row = n*BEAM+w, queries = x[n] - (iter+1)*cb[tuples[n][w][iter]],
//              ban stages of tuples[n][w][0..iter], output cand[row][0..3]
// Block: 128 threads (4 waves), 16 query rows per block.
// Codebook tiles double-buffered in LDS; next K-slice prefetched via the
// async-to-LDS engine while WMMA consumes the current slice.
// ---------------------------------------------------------------------------
__global__ __launch_bounds__(128)
void avq_dist_topk_kernel(const float* __restrict__ x,
                          const float* __restrict__ cb,
                          const float* __restrict__ cb2,
                          const int*   __restrict__ tin,
                          int*         __restrict__ outIdx,
                          int iter, int stage0)
{
  __shared__ float qtile[16 * QSTR];   // queries, full K=256
  __shared__ float btile[2 * BBUF];    // double-buffered 64x64 codebook slices
  __shared__ float dtile[16 * DSTR];   // raw dot products for scanning
  __shared__ float cbn[64];
  __shared__ float q2[16];
  __shared__ int   ban[16];

  const int tid  = threadIdx.x;
  const int lane = tid & 31;
  const int wv   = tid >> 5;
  const int rowBase = blockIdx.x * 16;

  if (tid < 16) q2[tid] = 0.f;
  __syncthreads();

  // ---- stage queries into LDS (8 threads per row, 32 floats each) ----
  {
    const int r   = tid >> 3;
    const int seg = tid & 7;
    const int row = rowBase + r;
    const float* xs;
    const float* cs = nullptr;
    float scale = 0.f;
    if (stage0) {
      xs = x + (size_t)row * DDIM;
      if (seg == 0) ban[r] = 0;
    } else {
      const int n = row >> 2, w = row & 3;
      const int ci = tin[(n * BEAM + w) * NSTAGE + iter];
      xs = x + (size_t)n * DDIM;
      cs = cb + (size_t)ci * DDIM;
      scale = (float)(iter + 1);
      if (seg == 0) {
        int m = 0;
        for (int s = 0; s <= iter; ++s) {
          int t = tin[(n * BEAM + w) * NSTAGE + s];
          m |= 1 << (t >> 10);
        }
        ban[r] = m;
      }
    }
    float part = 0.f;
    const int d0 = seg * 32;
#pragma unroll
    for (int i = 0; i < 8; ++i) {
      float4 xv = *(const float4*)(xs + d0 + i * 4);
      if (!stage0) {
        float4 cv = *(const float4*)(cs + d0 + i * 4);
        xv.x -= scale * cv.x; xv.y -= scale * cv.y;
        xv.z -= scale * cv.z; xv.w -= scale * cv.w;
      }
      float* qd = qtile + r * QSTR + d0 + i * 4;
      qd[0] = xv.x; qd[1] = xv.y; qd[2] = xv.z; qd[3] = xv.w;
      part += xv.x * xv.x + xv.y * xv.y + xv.z * xv.z + xv.w * xv.w;
    }
    atomicAdd(&q2[r], part);
  }
  __syncthreads();

  // ---- per-row running top-4 (valid in lanes tid<16) ----
  float d0t = INFF, d1t = INFF, d2t = INFF, d3t = INFF;
  int   i0t = 0,    i1t = 0,    i2t = 0,    i3t = 0;

  // WMMA fragment addressing per ISA layout:
  //   A 16x4 f32: lanes 0-15 -> M=lane, K={k,k+1}; lanes 16-31 -> M=lane-16, K={k+2,k+3}
  //   B 4x16 f32: lanes 0-15 -> N=lane, K={k,k+1}; lanes 16-31 -> N=lane-16, K={k+2,k+3}
  const int   ah   = (lane >= 16) ? 2 : 0;
  const int   al   = lane & 15;
  const float* qrow = qtile + al * QSTR;
  const int    c64  = wv * 16 + al;          // this wave's local column

  for (int colBase = 0; colBase < SK; colBase += 64) {
    if (tid < 64) cbn[tid] = cb2[colBase + tid];

    v8f acc = {0.f, 0.f, 0.f, 0.f, 0.f, 0.f, 0.f, 0.f};

    // prime the pipeline: slice 0 -> buffer 0
    avq_stage_slice(cb, colBase, 0, btile, tid);

    for (int s = 0; s < 4; ++s) {          // 4 K-slices of 64
      const float* cur = btile + (s & 1) * BBUF;
      if (s < 3)                            // prefetch next slice into other buf
        avq_stage_slice(cb, colBase, (s + 1) * 64, btile + ((s + 1) & 1) * BBUF, tid);
#if HAVE_ASYNC
      if (s < 3) __builtin_amdgcn_s_wait_asynccnt(8);  // current slice done,
      else       __builtin_amdgcn_s_wait_asynccnt(0);  // next may stay in flight
#endif
      __syncthreads();                      // slice visible to all waves

      const float* brow = cur + c64 * BSTR;
      const int ksq = s * 64;
#pragma unroll
      for (int k = 0; k < 64; k += 4) {
        v2f a = *(const v2f*)(qrow + ksq + k + ah);
        v2f b = *(const v2f*)(brow + k + ah);
        acc = __builtin_amdgcn_wmma_f32_16x16x4_f32(
            false, a, false, b, (short)0, acc, false, false);
      }
      __syncthreads();                      // all waves done with this buffer
    }

    // spill 16x16 dot tile: lane holds one column, 8 rows
    {
      const int rbase = (lane >= 16) ? 8 : 0;
#pragma unroll
      for (int r = 0; r < 8; ++r)
        dtile[(rbase + r) * DSTR + c64] = acc[r];
    }
    __syncthreads();

    // fused distance + ban + top-4 update (one lane per query row)
    if (tid < 16) {
      const float qq = q2[tid];
      const int   bm = ban[tid];
      for (int c = 0; c < 64; ++c) {
        const int gc = colBase + c;
        if (bm & (1 << (gc >> 10))) continue;
        const float d = qq - 2.f * dtile[tid * DSTR + c] + cbn[c];
        if (d < d3t) {
          d3t = d; i3t = gc;
          if (d3t < d2t) { float td = d2t; d2t = d3t; d3t = td; int ti = i2t; i2t = i3t; i3t = ti; }
          if (d2t < d1t) { float td = d1t; d1t = d2t; d2t = td; int ti = i1t; i1t = i2t; i2t = ti; }
          if (d1t < d0t) { float td = d0t; d0t = d1t; d1t = td; int ti = i0t; i0t = i1t; i1t = ti; }
        }
      }
    }
    __syncthreads();
  }

  if (tid < 16) {
    const int row = rowBase + tid;
    if (stage0) {
      int* tp = outIdx + (size_t)row * BEAM * NSTAGE;
      const int ids[4] = {i0t, i1t, i2t, i3t};
#pragma unroll
      for (int w = 0; w < BEAM; ++w) {
        tp[w * NSTAGE + 0] = ids[w];
        tp[w * NSTAGE + 1] = -1;
        tp[w * NSTAGE + 2] = -1;
        tp[w * NSTAGE + 3] = -1;
      }
    } else {
      int* cp = outIdx + (size_t)row * BEAM;
      cp[0] = i0t; cp[1] = i1t; cp[2] = i2t; cp[3] = i3t;
    }
  }
}

// ---------------------------------------------------------------------------
// Beam expansion + MSE selection. One block (256 thr) per sample.
// 16 candidates x 16 lanes (16 dims each).
// ---------------------------------------------------------------------------
__global__ __launch_bounds__(256)
void avq_beam_select_kernel(const float* __restrict__ x,
                            const float* __restrict__ cb,
                            const int*   __restrict__ tin,
                            const int*   __restrict__ cand,
                            int*         __restrict__ tout,
                            int*         __restrict__ best,
                            int iter)
{
  __shared__ int   prime[16][NSTAGE];
  __shared__ float part[16][16];
  __shared__ float mse[16];

  const int n   = blockIdx.x;
  const int tid = threadIdx.x;
  const int c   = tid >> 4;
  const int l   = tid & 15;

  if (tid < 16) {
    const int w = tid >> 2, j = tid & 3;
#pragma unroll
    for (int s = 0; s < NSTAGE; ++s)
      prime[tid][s] = (s <= iter) ? tin[(n * BEAM + w) * NSTAGE + s] : -1;
    prime[tid][iter + 1] = cand[(n * BEAM + w) * BEAM + j];
  }
  __syncthreads();

  float acc = 0.f;
  const int dbase = l * 16;
  for (int dv = 0; dv < 4; ++dv) {
    float4 q = {0.f, 0.f, 0.f, 0.f};
    for (int s = 0; s <= iter + 1; ++s) {
      const int idx = prime[c][s];
      float4 cv = *(const float4*)(cb + (size_t)idx * DDIM + dbase + dv * 4);
      q.x += cv.x; q.y += cv.y; q.z += cv.z; q.w += cv.w;
    }
    float4 xv = *(const float4*)(x + (size_t)n * DDIM + dbase + dv * 4);
    float ex = xv.x - q.x, ey = xv.y - q.y, ez = xv.z - q.z, ew = xv.w - q.w;
    acc += ex * ex + ey * ey + ez * ez + ew * ew;
  }
  part[c][l] = acc;
  __syncthreads();

  if (tid < 16) {
    float s = 0.f;
    for (int i = 0; i < 16; ++i) s += part[tid][i];
    mse[tid] = s;
  }
  __syncthreads();

  if (tid == 0) {
    if (iter < NSTAGE - 2) {
      unsigned taken = 0;
      for (int w = 0; w < BEAM; ++w) {
        float bd = INFF; int bi = 0;
        for (int cc = 0; cc < 16; ++cc) {
          if (taken & (1u << cc)) continue;
          if (mse[cc] < bd) { bd = mse[cc]; bi = cc; }
        }
        taken |= 1u << bi;
        for (int s = 0; s < NSTAGE; ++s)
          tout[(n * BEAM + w) * NSTAGE + s] = prime[bi][s];
      }
    } else {
      float bd = INFF; int bi = 0;
      for (int cc = 0; cc < 16; ++cc)
        if (mse[cc] < bd) { bd = mse[cc]; bi = cc; }
      for (int s = 0; s < NSTAGE; ++s)
        best[n * NSTAGE + s] = prime[bi][s];
    }
  }
}

// ---------------------------------------------------------------------------
// Final NSVQ output + used scatter. One block (256 thr = 256 dims) per sample.
// ---------------------------------------------------------------------------
__global__ __launch_bounds__(256)
void avq_nsvq_kernel(const float* __restrict__ x,
                     const float* __restrict__ rv,
                     const float* __restrict__ cb,
                     const int*   __restrict__ best,
                     float*       __restrict__ out,
                     float*       __restrict__ used)
{
  __shared__ float sh[256];
  __shared__ float sr[256];
  __shared__ float scaleS;

  const int n = blockIdx.x;
  const int d = threadIdx.x;

  float q = 0.f;
#pragma unroll
  for (int s = 0; s < NSTAGE; ++s) {
    const int idx = best[n * NSTAGE + s];
    if (idx >= 0) q += cb[(size_t)idx * DDIM + d];
  }
  const float xv  = x[(size_t)n * DDIM + d];
  const float rvv = rv[(size_t)n * DDIM + d];
  const float e = xv - q;
  sh[d] = e * e;
  sr[d] = rvv * rvv;
  __syncthreads();
  for (int st = 128; st > 0; st >>= 1) {
    if (d < st) { sh[d] += sh[d + st]; sr[d] += sr[d + st]; }
    __syncthreads();
  }
  if (d == 0) scaleS = sqrtf(sh[0]) / sqrtf(sr[0]) + EPSF;
  __syncthreads();
  out[(size_t)n * DDIM + d] = xv + scaleS * rvv;
  if (d < NSTAGE) {
    const int idx = best[n * NSTAGE + d];
    if (idx >= 0) used[idx] = 1.0f;   // benign races: all writers store 1.0
  }
}

// ---------------------------------------------------------------------------
extern "C" void kernel_launch(void* const* d_in, const int* in_sizes, int n_in,
                              void* d_out, int out_size, void* d_ws, size_t ws_size,
                              hipStream_t stream)
{
  (void)in_sizes; (void)n_in; (void)out_size; (void)ws_size;
  const float* x  = (const float*)d_in[0];
  const float* cb = (const float*)d_in[1];
  const float* rv = (const float*)d_in[2];
  // d_in[3] = train_mode (assumed 1 per reference setup)

  char* ws = (char*)d_ws;
  float* cb2  = (float*)(ws);                                 //  16 KB
  int*   tupA = (int*)  (ws + (16u << 10));                   // 256 KB
  int*   tupB = (int*)  (ws + (16u << 10) + (256u << 10));    // 256 KB
  int*   cand = (int*)  (ws + (16u << 10) + (512u << 10));    // 256 KB
  int*   best = (int*)  (ws + (16u << 10) + (768u << 10));    //  64 KB

  float* out  = (float*)d_out;
  float* used = out + (size_t)NPTS * DDIM;

  avq_cb2_kernel<<<SK / 256, 256, 0, stream>>>(cb, cb2);
  avq_zero_used_kernel<<<SK / 256, 256, 0, stream>>>(used);

  // Stage-0 seeding: top-4 of x vs all codebooks -> tuples slot 0
  avq_dist_topk_kernel<<<NPTS / 16, 128, 0, stream>>>(
      x, cb, cb2, nullptr, tupA, 0, /*stage0=*/1);

  int* tin = tupA;
  int* tout = tupB;
  for (int it = 0; it < NSTAGE - 1; ++it) {
    avq_dist_topk_kernel<<<(NPTS * BEAM) / 16, 128, 0, stream>>>(
        x, cb, cb2, tin, cand, it, /*stage0=*/0);
    avq_beam_select_kernel<<<NPTS, 256, 0, stream>>>(
        x, cb, tin, cand, tout, best, it);
    int* t = tin; tin = tout; tout = t;
  }

  avq_nsvq_kernel<<<NPTS, 256, 0, stream>>>(x, rv, cb, best, out, used);
}